// InterferenceModelingV1_33827162423519
// MI455X (gfx1250) — compile-verified
//
#include <hip/hip_runtime.h>
#include <hip/hip_bf16.h>

typedef __attribute__((ext_vector_type(2))) float v2f;
typedef __attribute__((ext_vector_type(8))) float v8f;

#define NODE_DIM   64
#define LEAKY      0.2f

// ---------------------------------------------------------------------------
// Kernel 1: projection GEMM  P[n, c] = sum_k X[n,k] * W[k,c],  c in [0,16)
//   W[k][c] = a_r_params[(c&7)*128 + (c>>3)*64 + k]
//   (c<8 -> a_u of relation c ; c>=8 -> a_v of relation c-8)
// One wave computes a 16x16 f32 tile with V_WMMA_F32_16X16X4_F32, K-loop of 16.
// 256 threads = 8 waves -> 128 rows per block.
// ---------------------------------------------------------------------------
__global__ void __launch_bounds__(256)
proj_gemm_wmma(const float* __restrict__ X, const float* __restrict__ A,
               float* __restrict__ P, int n_units) {
  const int lane = threadIdx.x & 31;
  const int wave = threadIdx.x >> 5;
  const int half = lane >> 4;       // 0: K pair {0,1}, 1: K pair {2,3}
  const int m    = lane & 15;       // A row within tile / B column within tile
  const int row_base = blockIdx.x * 128 + wave * 16;

  int row = row_base + m;
  if (row >= n_units) row = n_units - 1;          // clamp loads; stores guarded
  const float* __restrict__ xrow = X + (size_t)row * NODE_DIM;
  const float* __restrict__ wcol = A + (m & 7) * 128 + ((m >> 3) & 1) * 64;

  v8f c = {};
#pragma unroll
  for (int k0 = 0; k0 < NODE_DIM; k0 += 4) {
    const int k = k0 + half * 2;
    v2f a, b;
    a.x = xrow[k];     a.y = xrow[k + 1];   // A 16x4 frag (lane M, half -> K pair)
    b.x = wcol[k];     b.y = wcol[k + 1];   // B 4x16 frag (lane N, half -> K pair)
    c = __builtin_amdgcn_wmma_f32_16x16x4_f32(
        /*neg_a=*/false, a, /*neg_b=*/false, b,
        /*c_mod=*/(short)0, c, /*reuse_a=*/false, /*reuse_b=*/false);
  }

  // C/D layout: VGPR j, lanes 0-15 -> M=j, lanes 16-31 -> M=j+8; N = lane&15
#pragma unroll
  for (int j = 0; j < 8; ++j) {
    const int orow = row_base + j + half * 8;
    if (orow < n_units) P[(size_t)orow * 16 + m] = c[j];
  }
}

// ---------------------------------------------------------------------------
// Kernel 2: init per-node accumulators
// ---------------------------------------------------------------------------
__global__ void __launch_bounds__(256)
init_nodes(int* __restrict__ seg_max_bits, float* __restrict__ denom,
           float* __restrict__ gnum, int n) {
  const int i = blockIdx.x * blockDim.x + threadIdx.x;
  if (i < n) {
    seg_max_bits[i] = 0xFF800000;   // -inf
    denom[i] = 0.0f;
    gnum[i]  = 0.0f;
  }
}

// Ordered-bits float atomic max (monotone int mapping of IEEE754).
__device__ __forceinline__ void atomicMaxF32(float* addr, float v) {
  const int vi = __float_as_int(v);
  if (vi >= 0) atomicMax((int*)addr, vi);
  else         atomicMin((unsigned int*)addr, (unsigned int)vi);
}

__device__ __forceinline__ float edge_logit(const float* __restrict__ P,
                                            int s, int d, int r) {
  float e = P[(size_t)s * 16 + r] + P[(size_t)d * 16 + 8 + r];
  return (e > 0.0f) ? e : LEAKY * e;
}

// ---------------------------------------------------------------------------
// Kernel 3: per-edge logit + segment max over src
// ---------------------------------------------------------------------------
__global__ void __launch_bounds__(256)
edge_pass_max(const float* __restrict__ P, const int* __restrict__ src,
              const int* __restrict__ dst, const int* __restrict__ etype,
              float* __restrict__ seg_max, int n_edges) {
  const int i = blockIdx.x * blockDim.x + threadIdx.x;
  if (i >= n_edges) return;
  const int s = src[i], d = dst[i], r = etype[i];
  atomicMaxF32(&seg_max[s], edge_logit(P, s, d, r));
}

// ---------------------------------------------------------------------------
// Kernel 4: per-edge exp / segment sums (recompute e instead of spilling it)
// ---------------------------------------------------------------------------
__global__ void __launch_bounds__(256)
edge_pass_sum(const float* __restrict__ P, const int* __restrict__ src,
              const int* __restrict__ dst, const int* __restrict__ etype,
              const float* __restrict__ seg_max, const float* __restrict__ hat_t,
              float* __restrict__ denom, float* __restrict__ gnum, int n_edges) {
  const int i = blockIdx.x * blockDim.x + threadIdx.x;
  if (i >= n_edges) return;
  const int s = src[i], d = dst[i], r = etype[i];
  const float ex = __expf(edge_logit(P, s, d, r) - seg_max[s]);
  atomicAdd(&denom[s], ex);
  atomicAdd(&gnum[s], ex * hat_t[d]);
}

// ---------------------------------------------------------------------------
// Kernel 5: finalize  g_u = gnum / denom  (0 for edge-less nodes)
// ---------------------------------------------------------------------------
__global__ void __launch_bounds__(256)
finalize_nodes(const float* __restrict__ denom, const float* __restrict__ gnum,
               float* __restrict__ out, int n) {
  const int i = blockIdx.x * blockDim.x + threadIdx.x;
  if (i < n) {
    const float dn = denom[i];
    out[i] = (dn > 0.0f) ? (gnum[i] / dn) : 0.0f;
  }
}

// ---------------------------------------------------------------------------
extern "C" void kernel_launch(void* const* d_in, const int* in_sizes, int n_in,
                              void* d_out, int out_size, void* d_ws, size_t ws_size,
                              hipStream_t stream) {
  const float* X     = (const float*)d_in[0];   // (n_units, 64)
  const float* hat_t = (const float*)d_in[1];   // (n_units,)
  const float* A     = (const float*)d_in[2];   // (8, 128)
  const int*   eidx  = (const int*)d_in[3];     // (2, n_edges)
  const int*   etype = (const int*)d_in[4];     // (n_edges,)

  const int n_units = in_sizes[1];
  const int n_edges = in_sizes[4];
  const int* src = eidx;
  const int* dst = eidx + n_edges;

  // Workspace layout
  float* P        = (float*)d_ws;                         // n_units * 16
  float* seg_max  = P + (size_t)n_units * 16;             // n_units
  float* denom    = seg_max + n_units;                    // n_units
  float* gnum     = denom + n_units;                      // n_units
  float* out      = (float*)d_out;                        // n_units

  const int gemm_blocks = (n_units + 127) / 128;
  proj_gemm_wmma<<<gemm_blocks, 256, 0, stream>>>(X, A, P, n_units);

  const int node_blocks = (n_units + 255) / 256;
  init_nodes<<<node_blocks, 256, 0, stream>>>((int*)seg_max, denom, gnum, n_units);

  const int edge_blocks = (n_edges + 255) / 256;
  edge_pass_max<<<edge_blocks, 256, 0, stream>>>(P, src, dst, etype, seg_max, n_edges);
  edge_pass_sum<<<edge_blocks, 256, 0, stream>>>(P, src, dst, etype, seg_max, hat_t,
                                                 denom, gnum, n_edges);
  finalize_nodes<<<node_blocks, 256, 0, stream>>>(denom, gnum, out, n_units);
}